// EntityEncoder_81905026335238
// MI455X (gfx1250) — compile-verified
//
#include <hip/hip_runtime.h>

// ---------------------------------------------------------------------------
// EntityEncoder for MI455X (gfx1250): fp32 WMMA (V_WMMA_F32_16X16X4_F32)
// wave32, one wave per 16x16 output tile (or 16x128 strip for fused kernels).
// Async global->LDS staging (ASYNCcnt) for the relation-bucketed GEMM.
// ---------------------------------------------------------------------------

#define BB 32
#define LL 128
#define EC 48
#define NN 256
#define MM 512
#define DD 128
#define RR 100

typedef __attribute__((ext_vector_type(2))) float v2f;
typedef __attribute__((ext_vector_type(8))) float v8f;

__device__ __forceinline__ v8f wmma4(v2f a, v2f b, v8f c) {
  // D = A(16x4 f32) x B(4x16 f32) + C(16x16 f32)
  return __builtin_amdgcn_wmma_f32_16x16x4_f32(
      /*neg_a=*/false, a, /*neg_b=*/false, b,
      /*c_mod=*/(short)0, c, /*reuse_a=*/false, /*reuse_b=*/false);
}

// -------------------- 1) context entity hidden (out1) ----------------------
// out1[b,e,n] = relu( sum_k ceh[b,e,k]*mlp1_w[n,k] + mlp1_b[n] )
// ceh[b,e,k]  = mask * (k<D ? ctx_emb[b,pos,k] : ctx_out[b,pos,k-D])
__global__ __launch_bounds__(32)
void k_ctx_entity(const float* __restrict__ ctx_emb,
                  const float* __restrict__ ctx_out,
                  const int* __restrict__ pos, const int* __restrict__ cemask,
                  const float* __restrict__ w1, const float* __restrict__ b1,
                  float* __restrict__ out1) {
  int b = blockIdx.x, et = blockIdx.y, nt = blockIdx.z;
  int lane = threadIdx.x, r = lane & 15, hi = lane >> 4;
  int e = et * 16 + r;
  int p = pos[b * EC + e];
  float mk = (float)cemask[b * EC + e];
  const float* re = ctx_emb + ((size_t)b * LL + p) * DD;
  const float* ro = ctx_out + ((size_t)b * LL + p) * DD;
  const float* wn = w1 + (size_t)(nt * 16 + r) * (2 * DD);
  v8f acc = {};
  // first half of the concat: context_emb
  for (int k0 = 0; k0 < DD; k0 += 4) {
    int ka = k0 + 2 * hi;
    v2f a, bv;
    a.x = re[ka] * mk;
    a.y = re[ka + 1] * mk;
    bv.x = wn[ka];
    bv.y = wn[ka + 1];
    acc = wmma4(a, bv, acc);
  }
  // second half: context_outputs
  for (int k0 = DD; k0 < 2 * DD; k0 += 4) {
    int ka = k0 + 2 * hi;
    v2f a, bv;
    a.x = ro[ka - DD] * mk;
    a.y = ro[ka + 1 - DD] * mk;
    bv.x = wn[ka];
    bv.y = wn[ka + 1];
    acc = wmma4(a, bv, acc);
  }
  float bias = b1[nt * 16 + r];
#pragma unroll
  for (int j = 0; j < 8; ++j) {
    int m = j + 8 * hi;
    out1[((size_t)b * EC + et * 16 + m) * DD + nt * 16 + r] =
        fmaxf(acc[j] + bias, 0.0f);
  }
}

// -------------------- 2) entity embedding gather ----------------------------
__global__ void k_gather_emb(const int* __restrict__ entity,
                             const float* __restrict__ table,
                             float* __restrict__ emb) {
  int t = blockIdx.x * blockDim.x + threadIdx.x;  // B*N*(D/4) lanes
  int dv = t & 31, bn = t >> 5;
  if (bn >= BB * NN) return;
  int id = entity[bn];
  const float4* s = (const float4*)(table + (size_t)id * DD);
  ((float4*)(emb + (size_t)bn * DD))[dv] = s[dv];
}

// -------------------- 3) q = emb @ Wq^T + bq -------------------------------
__global__ __launch_bounds__(32)
void k_gemm_q(const float* __restrict__ A, const float* __restrict__ wq,
              const float* __restrict__ bq, float* __restrict__ out) {
  int mt = blockIdx.x, nt = blockIdx.y;
  int lane = threadIdx.x, r = lane & 15, hi = lane >> 4;
  const float* arow = A + (size_t)(mt * 16 + r) * DD;
  const float* brow = wq + (size_t)(nt * 16 + r) * DD;
  v8f acc = {};
  for (int k0 = 0; k0 < DD; k0 += 4) {
    int ka = k0 + 2 * hi;
    v2f a, bv;
    a.x = arow[ka];  a.y = arow[ka + 1];
    bv.x = brow[ka]; bv.y = brow[ka + 1];
    acc = wmma4(a, bv, acc);
  }
  float bias = bq[nt * 16 + r];
#pragma unroll
  for (int j = 0; j < 8; ++j) {
    int m = j + 8 * hi;
    out[(size_t)(mt * 16 + m) * DD + nt * 16 + r] = acc[j] + bias;
  }
}

// -------- 4) scores = q . ctx_emb, masked softmax over L (fused) -----------
__global__ __launch_bounds__(32)
void k_scores(const float* __restrict__ q, const float* __restrict__ ctx_emb,
              const int* __restrict__ cmask, float* __restrict__ attw) {
  __shared__ float s[16][LL];
  int b = blockIdx.x, ntl = blockIdx.y;  // 16 query rows per block
  int lane = threadIdx.x, r = lane & 15, hi = lane >> 4;
  const float* qrow = q + ((size_t)b * NN + ntl * 16 + r) * DD;
  for (int ct = 0; ct < 8; ++ct) {  // 8 column tiles over L
    v8f acc = {};
    const float* crow = ctx_emb + ((size_t)b * LL + ct * 16 + r) * DD;
    for (int k0 = 0; k0 < DD; k0 += 4) {
      int ka = k0 + 2 * hi;
      v2f a, bv;
      a.x = qrow[ka];  a.y = qrow[ka + 1];
      bv.x = crow[ka]; bv.y = crow[ka + 1];
      acc = wmma4(a, bv, acc);
    }
#pragma unroll
    for (int j = 0; j < 8; ++j) s[j + 8 * hi][ct * 16 + r] = acc[j];
  }
  __syncthreads();
  // row softmax: lane handles row r, half hi (cols hi*64 .. hi*64+63)
  int c0 = hi * 64;
  float mx = -1e30f;
  for (int c = c0; c < c0 + 64; ++c) {
    float v = (cmask[b * LL + c] > 0) ? s[r][c] : -1e9f;
    s[r][c] = v;
    mx = fmaxf(mx, v);
  }
  mx = fmaxf(mx, __shfl_xor(mx, 16));
  float sum = 0.0f;
  for (int c = c0; c < c0 + 64; ++c) {
    float e = __expf(s[r][c] - mx);
    s[r][c] = e;
    sum += e;
  }
  sum += __shfl_xor(sum, 16);
  float inv = 1.0f / sum;
  for (int c = c0; c < c0 + 64; ++c)
    attw[((size_t)b * NN + ntl * 16 + r) * LL + c] = s[r][c] * inv;
}

// -------------------- 5) entity_align = attw @ ctx_emb ---------------------
__global__ __launch_bounds__(32)
void k_align(const float* __restrict__ attw, const float* __restrict__ ctx_emb,
             float* __restrict__ out) {
  int b = blockIdx.x, mt = blockIdx.y, nt = blockIdx.z;
  int lane = threadIdx.x, r = lane & 15, hi = lane >> 4;
  const float* arow = attw + ((size_t)b * NN + mt * 16 + r) * LL;
  v8f acc = {};
  for (int k0 = 0; k0 < LL; k0 += 4) {
    int ka = k0 + 2 * hi;
    v2f a, bv;
    a.x = arow[ka]; a.y = arow[ka + 1];
    const float* c0 = ctx_emb + ((size_t)b * LL + ka) * DD + nt * 16 + r;
    bv.x = c0[0];   bv.y = c0[DD];
    acc = wmma4(a, bv, acc);
  }
#pragma unroll
  for (int j = 0; j < 8; ++j) {
    int m = j + 8 * hi;
    out[((size_t)b * NN + mt * 16 + m) * DD + nt * 16 + r] = acc[j];
  }
}

// --------- 6) entity_hidden = relu(concat(emb, align) @ mlp2^T + b2) -------
__global__ __launch_bounds__(32)
void k_ehid(const float* __restrict__ emb, const float* __restrict__ align,
            const float* __restrict__ w2, const float* __restrict__ b2,
            float* __restrict__ out) {
  int mt = blockIdx.x, nt = blockIdx.y;
  int lane = threadIdx.x, r = lane & 15, hi = lane >> 4;
  const float* e0 = emb + (size_t)(mt * 16 + r) * DD;
  const float* a0 = align + (size_t)(mt * 16 + r) * DD;
  const float* brow = w2 + (size_t)(nt * 16 + r) * (2 * DD);
  v8f acc = {};
  for (int k0 = 0; k0 < DD; k0 += 4) {  // concat half 1: emb
    int ka = k0 + 2 * hi;
    v2f a, bv;
    a.x = e0[ka];
    a.y = e0[ka + 1];
    bv.x = brow[ka]; bv.y = brow[ka + 1];
    acc = wmma4(a, bv, acc);
  }
  for (int k0 = DD; k0 < 2 * DD; k0 += 4) {  // concat half 2: align
    int ka = k0 + 2 * hi;
    v2f a, bv;
    a.x = a0[ka - DD];
    a.y = a0[ka + 1 - DD];
    bv.x = brow[ka]; bv.y = brow[ka + 1];
    acc = wmma4(a, bv, acc);
  }
  float bias = b2[nt * 16 + r];
#pragma unroll
  for (int j = 0; j < 8; ++j) {
    int m = j + 8 * hi;
    out[(size_t)(mt * 16 + m) * DD + nt * 16 + r] = fmaxf(acc[j] + bias, 0.0f);
  }
}

// -------------------- 7) kb_init gather (masked) ---------------------------
__global__ void k_kb_gather(const int* __restrict__ kbe,
                            const int* __restrict__ kbm,
                            const float* __restrict__ ehid,
                            float* __restrict__ kbinit) {
  int t = blockIdx.x * blockDim.x + threadIdx.x;
  int dv = t & 31, bm = t >> 5;
  if (bm >= BB * MM) return;
  int b = bm / MM;
  int id = kbe[bm];
  float mk = (float)kbm[bm];
  float4 v = ((const float4*)(ehid + ((size_t)b * NN + id) * DD))[dv];
  v.x *= mk; v.y *= mk; v.z *= mk; v.w *= mk;
  ((float4*)(kbinit + (size_t)bm * DD))[dv] = v;
}

// -------------------- 8) relation bucketing --------------------------------
__global__ void k_bucket_zero(int* counts) {
  int t = threadIdx.x;
  if (t < RR) counts[t] = 0;
}
__global__ void k_bucket_count(const int* __restrict__ col, int* counts) {
  int t = blockIdx.x * blockDim.x + threadIdx.x;
  if (t < BB * MM) atomicAdd(&counts[col[t]], 1);
}
__global__ void k_bucket_scan(const int* __restrict__ counts, int* offsets,
                              int* cursor) {
  if (threadIdx.x == 0) {
    int acc = 0;
    for (int r = 0; r < RR; ++r) {
      offsets[r] = acc;
      cursor[r] = acc;
      acc += counts[r];
    }
    offsets[RR] = acc;
  }
}
__global__ void k_bucket_scatter(const int* __restrict__ col, int* cursor,
                                 int* __restrict__ list) {
  int t = blockIdx.x * blockDim.x + threadIdx.x;
  if (t < BB * MM) {
    int idx = atomicAdd(&cursor[col[t]], 1);
    list[idx] = t;
  }
}

// ------- 9) kb_state[e,:] = W[rel(e)] @ kb_init[e,:]  (bucketed GEMM) ------
// Block (rel, tile16): edge vectors staged in LDS via GLOBAL_LOAD_ASYNC_TO_LDS
// (ASYNCcnt), then 8 col tiles x 32 K-steps of fp32 WMMA reading A from LDS.
__global__ __launch_bounds__(32)
void k_kb_state(const int* __restrict__ offsets, const int* __restrict__ list,
                const float* __restrict__ kbinit, const float* __restrict__ W,
                float* __restrict__ kbstate) {
  int rel = blockIdx.x, t = blockIdx.y;
  int start = offsets[rel], cnt = offsets[rel + 1] - start;
  if (t * 16 >= cnt) return;
  __shared__ float xs[16][DD];
  __shared__ int eid[16];
  int lane = threadIdx.x, r = lane & 15, hi = lane >> 4;
  const float* Wr = W + (size_t)rel * DD * DD;
  // warm a slice of the relation matrix (lowers to global_prefetch_b8)
  __builtin_prefetch(Wr + (size_t)lane * ((DD * DD) / 32), 0, 0);
  if (lane < 16) {
    int slot = t * 16 + lane;
    eid[lane] = (slot < cnt) ? list[start + slot] : -1;
  }
  __syncthreads();
  // async-stage the 16x128 A tile: one b128 per lane per row (16B/lane)
  unsigned lds_base = (unsigned)(size_t)(void*)&xs[0][0];
  for (int e = 0; e < 16; ++e) {
    int id = eid[e];  // uniform across the wave for each e
    unsigned ldsoff = lds_base + (unsigned)(e * DD * 4 + lane * 16);
    if (id >= 0) {
      const float* g = kbinit + (size_t)id * DD + lane * 4;
      asm volatile("global_load_async_to_lds_b128 %0, %1, off"
                   :: "v"(ldsoff), "v"(g)
                   : "memory");
    } else {
      ((float4*)xs[e])[lane] = make_float4(0.f, 0.f, 0.f, 0.f);  // pad row
    }
  }
  asm volatile("s_wait_asynccnt 0x0" ::: "memory");
  __syncthreads();
  for (int nt = 0; nt < 8; ++nt) {
    const float* brow = Wr + (size_t)(nt * 16 + r) * DD;  // B[k][n]=W[rel][n][k]
    v8f acc = {};
    for (int k0 = 0; k0 < DD; k0 += 4) {
      int ka = k0 + 2 * hi;
      v2f a, bv;
      a.x = xs[r][ka]; a.y = xs[r][ka + 1];
      bv.x = brow[ka]; bv.y = brow[ka + 1];
      acc = wmma4(a, bv, acc);
    }
#pragma unroll
    for (int j = 0; j < 8; ++j) {
      int m = j + 8 * hi;
      if (t * 16 + m < cnt)
        kbstate[(size_t)eid[m] * DD + nt * 16 + r] = acc[j];
    }
  }
}

// ---- 10) out2 = relu( nei@kb_state/deg + kb_init@W0^T )  (fully fused) ----
// One wave per 16x128 output strip; nei row read once; deg from A-loads.
__global__ __launch_bounds__(32)
void k_kb_final(const int* __restrict__ nei, const float* __restrict__ kbstate,
                const float* __restrict__ kbinit, const float* __restrict__ W0,
                float* __restrict__ out2) {
  int b = blockIdx.x, mt = blockIdx.y;
  int lane = threadIdx.x, r = lane & 15, hi = lane >> 4;
  __shared__ float degs[16];
  const int* nrow = nei + ((size_t)b * MM + mt * 16 + r) * MM;
  v8f acc[8] = {};
  float degp = 0.0f;
  for (int k0 = 0; k0 < MM; k0 += 4) {
    int ka = k0 + 2 * hi;
    v2f a;
    a.x = (float)nrow[ka];
    a.y = (float)nrow[ka + 1];
    degp += a.x + a.y;
    const float* s0 = kbstate + ((size_t)b * MM + ka) * DD + r;
#pragma unroll
    for (int nt = 0; nt < 8; ++nt) {
      v2f bv;
      bv.x = s0[nt * 16];
      bv.y = s0[nt * 16 + DD];
      acc[nt] = wmma4(a, bv, acc[nt]);
    }
  }
  degp += __shfl_xor(degp, 16);
  if (lane < 16) degs[lane] = fmaxf(degp, 1.0f);
  __syncthreads();
#pragma unroll
  for (int nt = 0; nt < 8; ++nt)
#pragma unroll
    for (int j = 0; j < 8; ++j) acc[nt][j] /= degs[j + 8 * hi];
  // kb_left accumulated into the same WMMA accumulators
  const float* arow = kbinit + ((size_t)b * MM + mt * 16 + r) * DD;
  for (int k0 = 0; k0 < DD; k0 += 4) {
    int ka = k0 + 2 * hi;
    v2f a;
    a.x = arow[ka];
    a.y = arow[ka + 1];
#pragma unroll
    for (int nt = 0; nt < 8; ++nt) {
      const float* brow = W0 + (size_t)(nt * 16 + r) * DD;
      v2f bv;
      bv.x = brow[ka];
      bv.y = brow[ka + 1];
      acc[nt] = wmma4(a, bv, acc[nt]);
    }
  }
#pragma unroll
  for (int nt = 0; nt < 8; ++nt)
#pragma unroll
    for (int j = 0; j < 8; ++j) {
      int m = j + 8 * hi;
      out2[((size_t)b * MM + mt * 16 + m) * DD + nt * 16 + r] =
          fmaxf(acc[nt][j], 0.0f);
    }
}

// ---------------------------------------------------------------------------
extern "C" void kernel_launch(void* const* d_in, const int* in_sizes, int n_in,
                              void* d_out, int out_size, void* d_ws,
                              size_t ws_size, hipStream_t stream) {
  const float* ctx_emb = (const float*)d_in[0];
  const float* ctx_out = (const float*)d_in[1];
  const int* cmask  = (const int*)d_in[3];
  const int* cepos  = (const int*)d_in[4];
  const int* cemask = (const int*)d_in[7];
  const int* entity = (const int*)d_in[8];
  const int* kbe    = (const int*)d_in[11];
  const int* kbm    = (const int*)d_in[13];
  const int* kbcol  = (const int*)d_in[14];
  const int* kbnei  = (const int*)d_in[15];
  const float* table = (const float*)d_in[17];
  const float* w1 = (const float*)d_in[18];
  const float* b1 = (const float*)d_in[19];
  const float* w2 = (const float*)d_in[20];
  const float* b2 = (const float*)d_in[21];
  const float* wq = (const float*)d_in[22];
  const float* bq = (const float*)d_in[23];
  const float* Wrel = (const float*)d_in[24];
  const float* W0 = (const float*)d_in[25];

  float* out1 = (float*)d_out;                  // [B,EC,D]
  float* out2 = out1 + (size_t)BB * EC * DD;    // [B,M,D]

  // workspace layout (floats then ints)
  float* emb     = (float*)d_ws;                          // [B*N, D]
  float* qbuf    = emb + (size_t)BB * NN * DD;            // [B*N, D]
  float* attw    = qbuf + (size_t)BB * NN * DD;           // [B*N, L]
  float* align   = attw + (size_t)BB * NN * LL;           // [B*N, D]
  float* ehid    = align + (size_t)BB * NN * DD;          // [B*N, D]
  float* kbinit  = ehid + (size_t)BB * NN * DD;           // [B*M, D]
  float* kbstate = kbinit + (size_t)BB * MM * DD;         // [B*M, D]
  int* counts  = (int*)(kbstate + (size_t)BB * MM * DD);  // [R]
  int* offsets = counts + RR;                             // [R+1]
  int* cursor  = offsets + RR + 1;                        // [R]
  int* list    = cursor + RR;                             // [B*M]

  // 1) out1
  k_ctx_entity<<<dim3(BB, EC / 16, DD / 16), 32, 0, stream>>>(
      ctx_emb, ctx_out, cepos, cemask, w1, b1, out1);
  // 2) embedding gather
  k_gather_emb<<<(BB * NN * 32) / 256, 256, 0, stream>>>(entity, table, emb);
  // 3) q
  k_gemm_q<<<dim3(BB * NN / 16, DD / 16), 32, 0, stream>>>(emb, wq, bq, qbuf);
  // 4) masked-softmax attention weights
  k_scores<<<dim3(BB, NN / 16), 32, 0, stream>>>(qbuf, ctx_emb, cmask, attw);
  // 5) align
  k_align<<<dim3(BB, NN / 16, DD / 16), 32, 0, stream>>>(attw, ctx_emb, align);
  // 6) entity_hidden
  k_ehid<<<dim3(BB * NN / 16, DD / 16), 32, 0, stream>>>(emb, align, w2, b2,
                                                         ehid);
  // 7) kb_init
  k_kb_gather<<<(BB * MM * 32) / 256, 256, 0, stream>>>(kbe, kbm, ehid,
                                                        kbinit);
  // 8) relation buckets
  k_bucket_zero<<<1, 128, 0, stream>>>(counts);
  k_bucket_count<<<(BB * MM) / 256, 256, 0, stream>>>(kbcol, counts);
  k_bucket_scan<<<1, 32, 0, stream>>>(counts, offsets, cursor);
  k_bucket_scatter<<<(BB * MM) / 256, 256, 0, stream>>>(kbcol, cursor, list);
  // 9) relation GEMM -> kb_state
  k_kb_state<<<dim3(RR, (BB * MM + 15) / 16), 32, 0, stream>>>(
      offsets, list, kbinit, Wrel, kbstate);
  // 10) fused neighbor aggregation + W0 GEMM + relu -> out2
  k_kb_final<<<dim3(BB, MM / 16), 32, 0, stream>>>(kbnei, kbstate, kbinit, W0,
                                                   out2);
}